// SelectiveSSM_47038481825920
// MI455X (gfx1250) — compile-verified
//
#include <hip/hip_runtime.h>
#include <hip/hip_bf16.h>
#include <math.h>
#include <stdint.h>

typedef __attribute__((ext_vector_type(2))) float v2f;
typedef __attribute__((ext_vector_type(8))) float v8f;
typedef __attribute__((ext_vector_type(4))) unsigned int u32x4;
typedef __attribute__((ext_vector_type(8))) int i32x8;
typedef __attribute__((ext_vector_type(4))) int i32x4;

#define D_INNER 1024
#define D_STATE 16
#define DT_RANK 64
#define KPROJ   (DT_RANK + 2 * D_STATE)   /* 96 */
#define BATCH   8
#define SEQLEN  2048
#define NTOK    (BATCH * SEQLEN)          /* 16384 */
#define DSLICE  256                        /* channels per scan block */
#define CHUNK   32                         /* timesteps staged per TDM tile */
#define NCHUNK  (SEQLEN / CHUNK)           /* 64 */

// ---------------------------------------------------------------------------
// Kernel 1: xp[16384 x 96] = X[16384 x 1024] * Wxp^T   (einsum 'btd,kd->btk')
// One wave computes a 16(M) x 96(N) tile via v_wmma_f32_16x16x4_f32.
// ---------------------------------------------------------------------------
__global__ __launch_bounds__(256) void ssm_proj_wmma(
    const float* __restrict__ X,
    const float* __restrict__ Wxp,
    float* __restrict__ xp)
{
    const int lane = threadIdx.x & 31;
    const int wave = threadIdx.x >> 5;
    const int m0   = (blockIdx.x * 8 + wave) * 16;
    const int r    = lane & 15;
    const int kh   = (lane >> 4) * 2;          // K sub-offset for this half-wave

    v8f acc[6] = {};                           // 6 N-tiles cover 96 columns

    const float* xrow = X + (size_t)(m0 + r) * D_INNER;

    for (int k = 0; k < D_INNER; k += 4) {
        const v2f a = *reinterpret_cast<const v2f*>(xrow + k + kh);
#pragma unroll
        for (int n = 0; n < 6; ++n) {
            const float* wrow = Wxp + (size_t)(n * 16 + r) * D_INNER + k + kh;
            const v2f b = *reinterpret_cast<const v2f*>(wrow);
            acc[n] = __builtin_amdgcn_wmma_f32_16x16x4_f32(
                false, a, false, b, (short)0, acc[n], false, false);
        }
    }

    const int mhi = (lane >> 4) * 8;
#pragma unroll
    for (int n = 0; n < 6; ++n) {
#pragma unroll
        for (int rr = 0; rr < 8; ++rr) {
            const int mm = m0 + rr + mhi;
            xp[(size_t)mm * KPROJ + n * 16 + r] = acc[n][rr];
        }
    }
}

// ---------------------------------------------------------------------------
// TDM helpers: build a 2D-tile D# (CDNA5 ISA) and issue
// TENSOR_LOAD_TO_LDS. rows x row_elems f32 tile, row_stride_elems between
// rows in global memory, packed contiguously into LDS at lds_byte_addr.
// This toolchain exposes the 6-arg builtin (extra int32x8 group, zero-filled).
// ---------------------------------------------------------------------------
typedef __attribute__((address_space(3))) void lds_void;

__device__ __forceinline__ uint32_t lds_offset(void* p)
{
    return (uint32_t)(uintptr_t)(lds_void*)p;     // LDS byte offset of shared var
}

__device__ __forceinline__ void tdm_load_tile_2d(
    const float* gsrc, uint32_t lds_byte_addr,
    uint32_t row_elems, uint32_t rows, uint32_t row_stride_elems)
{
    const uint64_t ga = (uint64_t)(uintptr_t)gsrc;

    u32x4 g0;
    g0.x = 1u;                                             // count=1 (valid), user D#
    g0.y = lds_byte_addr;                                  // lds_addr
    g0.z = (uint32_t)ga;                                   // global_addr[31:0]
    g0.w = (uint32_t)((ga >> 32) & 0x01FFFFFFull)          // global_addr[56:32]
           | (2u << 30);                                   // type = 2 ("image")

    i32x8 g1;
    g1[0] = (int)(2u << 16);                               // data_size = 4 bytes
    g1[1] = (int)(row_elems << 16);                        // tensor_dim0[15:0]
    g1[2] = (int)((row_elems >> 16) | (rows << 16));       // tdim0[31:16] | tdim1[15:0]
    g1[3] = (int)((rows >> 16) | (row_elems << 16));       // tdim1[31:16] | tile_dim0
    g1[4] = (int)rows;                                     // tile_dim1 | tile_dim2=0
    g1[5] = (int)row_stride_elems;                         // tensor_dim0_stride[31:0]
    g1[6] = 0;                                             // stride[47:32] | tdim1_stride lo
    g1[7] = 0;

    const i32x4 z4 = {0, 0, 0, 0};
    const i32x8 z8 = {0, 0, 0, 0, 0, 0, 0, 0};
    __builtin_amdgcn_tensor_load_to_lds(g0, g1, z4, z4, z8, 0);
}

// ---------------------------------------------------------------------------
// Kernel 2: fused dt-projection + softplus + selective scan, TDM-staged.
// One thread owns channel (b, d). Per CHUNK timesteps, wave 0 DMAs the next
// xp tile (CHUNK x 96) and x tile (CHUNK x 256) into LDS (double-buffered)
// while all waves compute the current chunk entirely out of LDS.
// Grid: 8 batches x 4 channel-blocks = 32 blocks of 256 threads.
// ---------------------------------------------------------------------------
__global__ __launch_bounds__(256) void ssm_scan(
    const float* __restrict__ X,
    const unsigned char* __restrict__ pad,     // bool mask (all-false in ref)
    const float* __restrict__ xp,
    const float* __restrict__ Wdt,
    const float* __restrict__ bdt,
    const float* __restrict__ logA,
    const float* __restrict__ Dp,
    float* __restrict__ Y)
{
    __shared__ float sxp[2][CHUNK][KPROJ];     // 24 KB
    __shared__ float sx [2][CHUNK][DSLICE];    // 64 KB

    const int b    = blockIdx.x >> 2;
    const int dblk = blockIdx.x & 3;
    const int d    = dblk * DSLICE + threadIdx.x;

    float w[DT_RANK];
#pragma unroll
    for (int i = 0; i < DT_RANK / 4; ++i) {
        const float4 v = reinterpret_cast<const float4*>(Wdt + (size_t)d * DT_RANK)[i];
        w[4 * i + 0] = v.x; w[4 * i + 1] = v.y;
        w[4 * i + 2] = v.z; w[4 * i + 3] = v.w;
    }

    float A[D_STATE], h[D_STATE];
#pragma unroll
    for (int n = 0; n < D_STATE; ++n) {
        A[n] = -__expf(logA[(size_t)d * D_STATE + n]);
        h[n] = 0.f;
    }
    const float bd = bdt[d];
    const float Dd = Dp[d];

    const float* xpb = xp + (size_t)b * SEQLEN * KPROJ;
    const float* xbs = X  + (size_t)b * SEQLEN * D_INNER + dblk * DSLICE;
    float*       yb  = Y  + (size_t)b * SEQLEN * D_INNER;

    const bool issuer = (threadIdx.x < 32);    // wave 0 owns the TDM queue

    if (issuer) {                              // prologue: stage chunk 0
        tdm_load_tile_2d(xpb, lds_offset(&sxp[0][0][0]), KPROJ, CHUNK, KPROJ);
        tdm_load_tile_2d(xbs, lds_offset(&sx[0][0][0]),  DSLICE, CHUNK, D_INNER);
    }

    for (int c = 0; c < NCHUNK; ++c) {
        if (issuer) {
            if (c + 1 < NCHUNK) {              // prefetch next chunk, then wait
                const int nb = (c + 1) & 1;    // for current chunk's 2 tiles
                tdm_load_tile_2d(xpb + (size_t)(c + 1) * CHUNK * KPROJ,
                                 lds_offset(&sxp[nb][0][0]), KPROJ, CHUNK, KPROJ);
                tdm_load_tile_2d(xbs + (size_t)(c + 1) * CHUNK * D_INNER,
                                 lds_offset(&sx[nb][0][0]), DSLICE, CHUNK, D_INNER);
                __builtin_amdgcn_s_wait_tensorcnt(2);
            } else {
                __builtin_amdgcn_s_wait_tensorcnt(0);
            }
        }
        __syncthreads();                       // chunk c resident for all waves

        const int cb = c & 1;
        for (int tt = 0; tt < CHUNK; ++tt) {
            const int t = c * CHUNK + tt;

            // dt = softplus(xp[b,t,:64] . w + bdt)   (LDS broadcast reads)
            float dtp = bd;
#pragma unroll
            for (int rr = 0; rr < DT_RANK; ++rr)
                dtp = fmaf(sxp[cb][tt][rr], w[rr], dtp);
            const float dt = (dtp > 20.f) ? dtp : log1pf(__expf(dtp));

            const float xv = sx[cb][tt][threadIdx.x];
            const float m  = pad[(size_t)b * SEQLEN + t] ? 0.f : 1.f;
            const float dx = dt * xv;

            float y = 0.f;
#pragma unroll
            for (int n = 0; n < D_STATE; ++n) {
                const float dA = __expf(A[n] * dt);
                const float hn = fmaf(dA, h[n], dx * sxp[cb][tt][DT_RANK + n]);      // B_t
                y    = fmaf(hn, sxp[cb][tt][DT_RANK + D_STATE + n], y);              // C_t
                h[n] = hn * m;
            }
            y = (y + Dd * xv) * m;
            yb[(size_t)t * D_INNER + d] = y;
        }
        __syncthreads();                       // all waves done with buf (c&1)
    }
}

// ---------------------------------------------------------------------------
// Host launcher. Workspace layout: xp = 16384 * 96 f32 = 6.29 MB.
// ---------------------------------------------------------------------------
extern "C" void kernel_launch(void* const* d_in, const int* in_sizes, int n_in,
                              void* d_out, int out_size, void* d_ws, size_t ws_size,
                              hipStream_t stream)
{
    const float*         x    = (const float*)d_in[0];
    const unsigned char* pad  = (const unsigned char*)d_in[1];   // bool mask
    const float*         Wxp  = (const float*)d_in[2];
    const float*         Wdt  = (const float*)d_in[3];
    const float*         bdt  = (const float*)d_in[4];
    const float*         logA = (const float*)d_in[5];
    const float*         Dp   = (const float*)d_in[6];
    float*               out  = (float*)d_out;
    float*               xp   = (float*)d_ws;                    // (16384, 96)

    (void)in_sizes; (void)n_in; (void)out_size; (void)ws_size;

    // 128 blocks x 256 threads: 8 waves/block, one 16x96 output tile per wave
    ssm_proj_wmma<<<NTOK / (16 * 8), 256, 0, stream>>>(x, Wxp, xp);

    // 32 blocks x 256 threads: one (b, d) channel per thread, T sequential,
    // TDM double-buffered LDS staging of xp and x tiles
    ssm_scan<<<BATCH * 4, 256, 0, stream>>>(x, pad, xp, Wdt, bdt, logA, Dp, out);
}